// BahdanauAttention_28475633173164
// MI455X (gfx1250) — compile-verified
//
#include <hip/hip_runtime.h>

// Problem constants (match reference)
#define Hh   128
#define Bb   8
#define TQq  512
#define TKk  512
#define NROWS (Bb * TQq)           // 4096 rows per projection
#define CTX_ELEMS (Bb * TQq * Hh)  // 524288 contexts elements (first output)

typedef __attribute__((ext_vector_type(2))) float v2f;
typedef __attribute__((ext_vector_type(8))) float v8f;

__device__ __forceinline__ float fast_tanhf(float x) {
#if __has_builtin(__builtin_amdgcn_tanhf)
    return __builtin_amdgcn_tanhf(x);           // gfx1250 v_tanh_f32
#else
    float e = __expf(2.0f * x);                 // v_exp_f32
    return 1.0f - 2.0f * __builtin_amdgcn_rcpf(e + 1.0f);
#endif
}

// ---------------------------------------------------------------------------
// Kernel 1: fused q/k projections  Y = X @ W^T + bias   (fp32 WMMA 16x16x4)
// block = (32,8) -> 8 waves, each wave owns one 16x16 output tile.
// grid  = (NROWS/16/8, H/16, 2)   z: 0 = q-projection, 1 = k-projection
// ---------------------------------------------------------------------------
__global__ void bahdanau_proj_wmma(const float* __restrict__ queries,
                                   const float* __restrict__ keys,
                                   const float* __restrict__ Wa_w,
                                   const float* __restrict__ Wa_b,
                                   const float* __restrict__ Ua_w,
                                   const float* __restrict__ Ua_b,
                                   float* __restrict__ qproj,
                                   float* __restrict__ kproj) {
    const int lane = threadIdx.x;        // 0..31
    const int wv   = threadIdx.y;        // 0..7
    const int half = lane >> 4;          // 0 or 1
    const int m    = lane & 15;

    const int which = blockIdx.z;
    const float* __restrict__ X    = which ? keys  : queries;   // (4096,128)
    const float* __restrict__ W    = which ? Ua_w  : Wa_w;      // (128,128) [out,in]
    const float* __restrict__ bias = which ? Ua_b  : Wa_b;      // (128)
    float* __restrict__ Y          = which ? kproj : qproj;     // (4096,128)

    const int row0 = (blockIdx.x * 8 + wv) * 16;
    const int col0 = blockIdx.y * 16;

    v8f acc = {};
#pragma unroll
    for (int k0 = 0; k0 < Hh; k0 += 4) {
        const int kk = k0 + 2 * half;
        v2f a, bm;
        // A = X tile (16x4, row-major M x K)
        a.x  = X[(row0 + m) * Hh + kk];
        a.y  = X[(row0 + m) * Hh + kk + 1];
        // B = W^T tile (4x16):  B[k][n] = W[col0+n][k]
        bm.x = W[(col0 + m) * Hh + kk];
        bm.y = W[(col0 + m) * Hh + kk + 1];
        acc = __builtin_amdgcn_wmma_f32_16x16x4_f32(false, a, false, bm,
                                                    (short)0, acc, false, false);
    }
    const float bj = bias[col0 + m];
#pragma unroll
    for (int g = 0; g < 8; ++g) {
        const int r = row0 + g + 8 * half;        // C layout: VGPR g -> M=g / g+8
        Y[r * Hh + col0 + m] = acc[g] + bj;
    }
}

// ---------------------------------------------------------------------------
// Kernel 2: fused scores + softmax.  One 256-thread block per (b, q) row.
//  score[k] = sum_h Va[h]*tanh(q[h]+k[h]) + Va_b ; weights = softmax_k(score)
// Lane L owns h = 4L..4L+3 (float4, coalesced); wave reduce via __shfl_xor.
// ---------------------------------------------------------------------------
__global__ void bahdanau_scores_softmax(const float* __restrict__ qproj,
                                        const float* __restrict__ kproj,
                                        const float* __restrict__ Va_w,
                                        const float* __restrict__ Va_b,
                                        float* __restrict__ weights) {
    __shared__ float s_scores[TKk];
    __shared__ float s_max[8];
    __shared__ float s_sum[8];

    const int b    = blockIdx.y;
    const int q    = blockIdx.x;
    const int t    = threadIdx.x;        // 0..255
    const int lane = t & 31;
    const int wv   = t >> 5;             // 0..7

    const float4 qv = *(const float4*)(qproj + ((size_t)(b * TQq + q)) * Hh + 4 * lane);
    const float4 va = *(const float4*)(Va_w + 4 * lane);
    const float  vb = Va_b[0];

    for (int k = wv; k < TKk; k += 8) {
        const float4 kv = *(const float4*)(kproj + ((size_t)(b * TKk + k)) * Hh + 4 * lane);
        float p  = va.x * fast_tanhf(qv.x + kv.x);
        p       += va.y * fast_tanhf(qv.y + kv.y);
        p       += va.z * fast_tanhf(qv.z + kv.z);
        p       += va.w * fast_tanhf(qv.w + kv.w);
#pragma unroll
        for (int off = 16; off >= 1; off >>= 1) p += __shfl_xor(p, off, 32);
        if (lane == 0) s_scores[k] = p + vb;
    }
    __syncthreads();

    // softmax over 512 scores: each thread handles 2 elements
    const float s0 = s_scores[t];
    const float s1 = s_scores[t + 256];
    float mx = fmaxf(s0, s1);
#pragma unroll
    for (int off = 16; off >= 1; off >>= 1) mx = fmaxf(mx, __shfl_xor(mx, off, 32));
    if (lane == 0) s_max[wv] = mx;
    __syncthreads();
    float bm = s_max[0];
#pragma unroll
    for (int w = 1; w < 8; ++w) bm = fmaxf(bm, s_max[w]);

    const float e0 = __expf(s0 - bm);
    const float e1 = __expf(s1 - bm);
    float sm = e0 + e1;
#pragma unroll
    for (int off = 16; off >= 1; off >>= 1) sm += __shfl_xor(sm, off, 32);
    if (lane == 0) s_sum[wv] = sm;
    __syncthreads();
    float tot = 0.0f;
#pragma unroll
    for (int w = 0; w < 8; ++w) tot += s_sum[w];
    const float inv = __builtin_amdgcn_rcpf(tot);

    float* __restrict__ wrow = weights + ((size_t)(b * TQq + q)) * TKk;
    wrow[t]       = e0 * inv;
    wrow[t + 256] = e1 * inv;
}

// ---------------------------------------------------------------------------
// Kernel 3: contexts = weights @ keys  (per batch 512x512 @ 512x128, fp32 WMMA)
// block = (32,8): wave wv owns h-tile wv; grid = (TQ/16, B)
// ---------------------------------------------------------------------------
__global__ void bahdanau_context_wmma(const float* __restrict__ weights,
                                      const float* __restrict__ keys,
                                      float* __restrict__ contexts) {
    const int lane = threadIdx.x;
    const int wv   = threadIdx.y;        // h tile index 0..7
    const int half = lane >> 4;
    const int m    = lane & 15;
    const int b    = blockIdx.y;
    const int q0   = blockIdx.x * 16;
    const int h0   = wv * 16;

    const float* __restrict__ Wt = weights + (size_t)b * TQq * TKk;  // (512,512)
    const float* __restrict__ Km = keys    + (size_t)b * TKk * Hh;   // (512,128)

    v8f acc = {};
    for (int k0 = 0; k0 < TKk; k0 += 4) {
        const int kk = k0 + 2 * half;
        v2f a, bm;
        a.x  = Wt[(q0 + m) * TKk + kk];        // A = weights tile (16x4)
        a.y  = Wt[(q0 + m) * TKk + kk + 1];
        bm.x = Km[kk * Hh + h0 + m];           // B = keys tile (4x16)
        bm.y = Km[(kk + 1) * Hh + h0 + m];
        acc = __builtin_amdgcn_wmma_f32_16x16x4_f32(false, a, false, bm,
                                                    (short)0, acc, false, false);
    }
#pragma unroll
    for (int g = 0; g < 8; ++g) {
        const int r = q0 + g + 8 * half;
        contexts[((size_t)(b * TQq) + r) * Hh + h0 + m] = acc[g];
    }
}

// ---------------------------------------------------------------------------
extern "C" void kernel_launch(void* const* d_in, const int* in_sizes, int n_in,
                              void* d_out, int out_size, void* d_ws, size_t ws_size,
                              hipStream_t stream) {
    (void)in_sizes; (void)n_in; (void)out_size; (void)ws_size;

    const float* queries = (const float*)d_in[0];
    const float* keys    = (const float*)d_in[1];
    const float* Wa_w    = (const float*)d_in[2];
    const float* Wa_b    = (const float*)d_in[3];
    const float* Ua_w    = (const float*)d_in[4];
    const float* Ua_b    = (const float*)d_in[5];
    const float* Va_w    = (const float*)d_in[6];
    const float* Va_b    = (const float*)d_in[7];

    float* contexts = (float*)d_out;                 // (B,TQ,H)
    float* weights  = (float*)d_out + CTX_ELEMS;     // (B,TQ,TK)

    float* qproj = (float*)d_ws;                     // (B*TQ, H)
    float* kproj = qproj + (size_t)NROWS * Hh;       // (B*TK, H)

    // K1: projections
    dim3 blk1(32, 8);
    dim3 grd1(NROWS / 16 / 8, Hh / 16, 2);
    bahdanau_proj_wmma<<<grd1, blk1, 0, stream>>>(queries, keys, Wa_w, Wa_b,
                                                  Ua_w, Ua_b, qproj, kproj);

    // K2: fused tanh-scores + softmax
    dim3 grd2(TQq, Bb);
    bahdanau_scores_softmax<<<grd2, dim3(256), 0, stream>>>(qproj, kproj,
                                                            Va_w, Va_b, weights);

    // K3: contexts = weights @ keys
    dim3 grd3(TQq / 16, Bb);
    bahdanau_context_wmma<<<grd3, dim3(32, 8), 0, stream>>>(weights, keys, contexts);
}